// JKNet_32066225832232
// MI455X (gfx1250) — compile-verified
//
#include <hip/hip_runtime.h>
#include <hip/hip_bf16.h>
#include <math.h>

// ---------------------------------------------------------------------------
// JKNet (6-layer GCN + JumpingKnowledge-max) for MI455X / gfx1250.
//  - GEMMs use V_WMMA_F32_16X16X4_F32 (exact fp32, wave32 fragment layout),
//    with B staged through LDS in pair-interleaved layout so each B fragment
//    is a single ds_load_b64 instead of two scalar global loads.
//  - Edge aggregation uses global_atomic_add_f32 (agg fits in 192MB L2).
// ---------------------------------------------------------------------------

typedef __attribute__((ext_vector_type(2))) float v2f;
typedef __attribute__((ext_vector_type(8))) float v8f;

#define HID 64
#define NCLS 40
#define BK 64                    // k-chunk staged in LDS per barrier interval

// ---- degree / norm prep ----------------------------------------------------

__global__ void init_deg_kernel(float* __restrict__ deg, int N) {
    int i = blockIdx.x * blockDim.x + threadIdx.x;
    if (i < N) deg[i] = 1.0f;                    // self-loop contributes 1
}

__global__ void edge_deg_kernel(const int* __restrict__ ei, float* __restrict__ deg, int E) {
    int e = blockIdx.x * blockDim.x + threadIdx.x;
    if (e < E) atomicAdd(&deg[ei[E + e]], 1.0f); // dst row of edge_index
}

__global__ void rsqrt_kernel(float* __restrict__ deg, int N) {
    int i = blockIdx.x * blockDim.x + threadIdx.x;
    if (i < N) deg[i] = rsqrtf(deg[i]);          // deg >= 1 always
}

__global__ void norm_kernel(const int* __restrict__ ei, const float* __restrict__ dinv,
                            float* __restrict__ norm, int E) {
    int e = blockIdx.x * blockDim.x + threadIdx.x;
    if (e < E) norm[e] = dinv[ei[e]] * dinv[ei[E + e]];
}

// ---- WMMA GEMM: C[M x 64] = A[M x K] * B[K x 64], row-major --------------
// One wave computes a 16(M) x 64(N) strip: 4 accumulators sharing one A frag.
// f32 16x16x4 fragment layout (ISA 7.12.2):
//   A: lanes 0-15 = rows M=0..15 with (K=0,K=1); lanes 16-31 same rows (K=2,K=3)
//   B: lanes 0-15 = cols N=0..15 with (K=0,K=1); lanes 16-31 same cols (K=2,K=3)
//   C/D: VGPR r -> lanes 0-15: (M=r, N=lane); lanes 16-31: (M=8+r, N=lane-16)
// B staged in LDS as pairs Bp[p][n] = (B[2p][n], B[2p+1][n]) so a lane's
// fragment = Bp[(k>>1)+(lane>>4)][n] -> one ds_load_b64.

__global__ __launch_bounds__(128)
void gemm64_wmma_kernel(const float* __restrict__ A, const float* __restrict__ B,
                        float* __restrict__ C, int M, int K) {
    __shared__ v2f Bp[(BK / 2) * HID];           // 32*64 pairs = 16KB

    const int tid   = threadIdx.x;
    const int lane  = tid & 31;
    const int wave  = tid >> 5;
    const int mtile = blockIdx.x * 4 + wave;
    const bool active = (mtile * 16 < M);        // tail waves still barrier

    const int mrow  = lane & 15;
    const int khalf = (lane >> 4) << 1;          // 0 or 2
    const int n0    = lane & 15;
    const int phalf = lane >> 4;                 // pair offset 0 or 1

    const float* __restrict__ arow =
        A + (active ? (size_t)(mtile * 16 + mrow) * K + khalf : (size_t)0);

    v8f c0 = {}; v8f c1 = {}; v8f c2 = {}; v8f c3 = {};

    for (int kc = 0; kc < K; kc += BK) {
        __syncthreads();                         // protect prior chunk reads
        // cooperative pair-interleaved fill of B chunk [kc, kc+BK)
        for (int idx = tid; idx < (BK / 2) * HID; idx += 128) {
            int p = idx >> 6, n = idx & 63;
            const float* __restrict__ brow = B + (size_t)(kc + p * 2) * HID + n;
            v2f bp; bp[0] = brow[0]; bp[1] = brow[HID];
            Bp[idx] = bp;
        }
        __syncthreads();

        if (active) {
            #pragma unroll 4
            for (int k = 0; k < BK; k += 4) {
                v2f a = *(const v2f*)(arow + kc + k);      // A[m][kc+k+khalf..+1]
                const v2f* __restrict__ bq = &Bp[((k >> 1) + phalf) * HID];
                v2f b0 = bq[n0 +  0];
                v2f b1 = bq[n0 + 16];
                v2f b2 = bq[n0 + 32];
                v2f b3 = bq[n0 + 48];
                c0 = __builtin_amdgcn_wmma_f32_16x16x4_f32(false, a, false, b0, (short)0, c0, false, false);
                c1 = __builtin_amdgcn_wmma_f32_16x16x4_f32(false, a, false, b1, (short)0, c1, false, false);
                c2 = __builtin_amdgcn_wmma_f32_16x16x4_f32(false, a, false, b2, (short)0, c2, false, false);
                c3 = __builtin_amdgcn_wmma_f32_16x16x4_f32(false, a, false, b3, (short)0, c3, false, false);
            }
        }
    }

    if (active) {
        const int mo = (lane >> 4) * 8;          // row offset 0 or 8
        float* __restrict__ crow = C + (size_t)(mtile * 16 + mo) * HID + n0;
        #pragma unroll
        for (int r = 0; r < 8; ++r) {
            crow[(size_t)r * HID +  0] = c0[r];
            crow[(size_t)r * HID + 16] = c1[r];
            crow[(size_t)r * HID + 32] = c2[r];
            crow[(size_t)r * HID + 48] = c3[r];
        }
    }
}

// ---- aggregation -----------------------------------------------------------

// agg[n][f] = b[f] + dinv[n]^2 * t[n][f]   (self-loop term + bias)
__global__ void selfloop_bias_kernel(const float* __restrict__ t, const float* __restrict__ dinv,
                                     const float* __restrict__ b, float* __restrict__ agg,
                                     int total) {
    int i = blockIdx.x * blockDim.x + threadIdx.x;
    if (i < total) {
        int n = i >> 6;
        float di = dinv[n];
        agg[i] = b[i & 63] + di * di * t[i];
    }
}

// one wave per edge, 2 features per lane; agg (25.6MB) is L2-resident
__global__ void scatter_kernel(const int* __restrict__ ei, const float* __restrict__ norm,
                               const float* __restrict__ t, float* __restrict__ agg, int E) {
    int idx  = blockIdx.x * blockDim.x + threadIdx.x;
    int e    = idx >> 5;
    int lane = idx & 31;
    if (e < E) {
        int   s = ei[e];
        int   d = ei[E + e];
        float w = norm[e];
        v2f v = *(const v2f*)(t + (size_t)s * HID + lane * 2);
        float* dp = agg + (size_t)d * HID + lane * 2;
        atomicAdd(dp + 0, w * v[0]);
        atomicAdd(dp + 1, w * v[1]);
    }
}

__global__ void relu_jk_kernel(const float* __restrict__ agg, float* __restrict__ h,
                               float* __restrict__ jk, int total, int first) {
    int i = blockIdx.x * blockDim.x + threadIdx.x;
    if (i < total) {
        float v = fmaxf(agg[i], 0.0f);
        h[i]  = v;
        jk[i] = first ? v : fmaxf(jk[i], v);
    }
}

// ---- FC + log_softmax ------------------------------------------------------
// one thread per node: 40 logits in VGPRs; W reads are thread-uniform.

__global__ __launch_bounds__(256)
void fc_logsoftmax_kernel(const float* __restrict__ jk, const float* __restrict__ W,
                          const float* __restrict__ b, float* __restrict__ out, int N) {
    int n = blockIdx.x * blockDim.x + threadIdx.x;
    if (n >= N) return;

    float logit[NCLS];
    #pragma unroll
    for (int c = 0; c < NCLS; ++c) logit[c] = b[c];

    const float* __restrict__ row = jk + (size_t)n * HID;
    for (int f = 0; f < HID; ++f) {
        float v = row[f];
        const float* __restrict__ wr = W + f * NCLS;
        #pragma unroll
        for (int c = 0; c < NCLS; ++c) logit[c] = fmaf(v, wr[c], logit[c]);
    }

    float m = logit[0];
    #pragma unroll
    for (int c = 1; c < NCLS; ++c) m = fmaxf(m, logit[c]);
    float s = 0.0f;
    #pragma unroll
    for (int c = 0; c < NCLS; ++c) s += expf(logit[c] - m);
    float ls = logf(s);

    float* __restrict__ o = out + (size_t)n * NCLS;
    #pragma unroll
    for (int c = 0; c < NCLS; ++c) o[c] = logit[c] - m - ls;
}

// ---------------------------------------------------------------------------

extern "C" void kernel_launch(void* const* d_in, const int* in_sizes, int n_in,
                              void* d_out, int out_size, void* d_ws, size_t ws_size,
                              hipStream_t stream) {
    const float* x   = (const float*)d_in[0];   // [N, 512]
    const int*   ei  = (const int*)  d_in[1];   // [2, E]
    const float* W0  = (const float*)d_in[2];   // [512, 64]
    const float* b0  = (const float*)d_in[3];   // [64]
    const float* Wr  = (const float*)d_in[4];   // [5, 64, 64]
    const float* br  = (const float*)d_in[5];   // [5, 64]
    const float* fcW = (const float*)d_in[6];   // [64, 40]
    const float* fcb = (const float*)d_in[7];   // [40]
    float*       out = (float*)d_out;           // [N, 40]

    const int IN_FEAT = 512;
    const int N = in_sizes[0] / IN_FEAT;        // 100000
    const int E = in_sizes[1] / 2;              // 1600000
    const int NLAYERS = 1 + in_sizes[5] / HID;  // 6

    // workspace layout (floats)
    float* ws = (float*)d_ws;
    size_t Np = ((size_t)N + 255) & ~(size_t)255;
    size_t Ep = ((size_t)E + 255) & ~(size_t)255;
    float* dinv = ws;                           // [N]
    float* norm = dinv + Np;                    // [E]
    float* t    = norm + Ep;                    // [N,64] GEMM output
    float* agg  = t    + (size_t)N * HID;       // [N,64] scatter accumulator
    float* h    = agg  + (size_t)N * HID;       // [N,64] layer activations
    float* jk   = h    + (size_t)N * HID;       // [N,64] JK running max

    const int B = 256;
    const int total = N * HID;

    // gcn_norm
    init_deg_kernel<<<(N + B - 1) / B, B, 0, stream>>>(dinv, N);
    edge_deg_kernel<<<(E + B - 1) / B, B, 0, stream>>>(ei, dinv, E);
    rsqrt_kernel<<<(N + B - 1) / B, B, 0, stream>>>(dinv, N);
    norm_kernel<<<(E + B - 1) / B, B, 0, stream>>>(ei, dinv, norm, E);

    const int mtiles   = (N + 15) / 16;
    const int gemmGrid = (mtiles + 3) / 4;      // 4 waves (16-row tiles) per block
    const int scatGrid = (int)(((size_t)E * 32 + B - 1) / B);

    for (int l = 0; l < NLAYERS; ++l) {
        const float* A  = (l == 0) ? x  : h;
        const int    K  = (l == 0) ? IN_FEAT : HID;
        const float* W  = (l == 0) ? W0 : Wr + (size_t)(l - 1) * HID * HID;
        const float* bb = (l == 0) ? b0 : br + (size_t)(l - 1) * HID;

        gemm64_wmma_kernel<<<gemmGrid, 128, 0, stream>>>(A, W, t, N, K);
        selfloop_bias_kernel<<<(total + B - 1) / B, B, 0, stream>>>(t, dinv, bb, agg, total);
        scatter_kernel<<<scatGrid, B, 0, stream>>>(ei, norm, t, agg, E);
        relu_jk_kernel<<<(total + B - 1) / B, B, 0, stream>>>(agg, h, jk, total, l == 0 ? 1 : 0);
    }

    fc_logsoftmax_kernel<<<(N + B - 1) / B, B, 0, stream>>>(jk, fcW, fcb, out, N);
}